// nc_GNN_DropMessage_5841155523227
// MI455X (gfx1250) — compile-verified
//
#include <hip/hip_runtime.h>
#include <hip/hip_bf16.h>

typedef float v2f __attribute__((ext_vector_type(2)));
typedef float v8f __attribute__((ext_vector_type(8)));

#define GCN_N 50000
#define GCN_K 128

static __device__ __forceinline__ void atomic_add_f32(float* p, float v) {
    __hip_atomic_fetch_add(p, v, __ATOMIC_RELAXED, __HIP_MEMORY_SCOPE_AGENT);
}

// ---------------- degree / dinv ----------------

__global__ void fill_one_kernel(float* __restrict__ d, int n) {
    int i = blockIdx.x * blockDim.x + threadIdx.x;
    if (i < n) d[i] = 1.0f;  // self-loop contributes 1 to every degree
}

__global__ void degree_scatter_kernel(const int* __restrict__ col, float* __restrict__ deg, int e) {
    int i = blockIdx.x * blockDim.x + threadIdx.x;
    if (i < e) atomic_add_f32(&deg[col[i]], 1.0f);
}

__global__ void rsqrt_kernel(float* __restrict__ d, int n) {
    int i = blockIdx.x * blockDim.x + threadIdx.x;
    if (i < n) d[i] = rsqrtf(d[i]);  // deg >= 1 always
}

// ---------------- dense GEMM via V_WMMA_F32_16X16X4_F32 ----------------
// out[M, NC] = A[M, 128] x W[128, NC]; one wave computes a 16 x NC strip.
// A-frag (16x4 f32, 2 VGPR): lane&15 = M row, (lane>>4)*2 = K pair base.
// B-frag mirrored: lane&15 = N col. C/D: VGPR r -> row r (+8 upper half-wave).

template <int NC>
__global__ void gemm_wmma_f32(const float* __restrict__ A, const float* __restrict__ W,
                              float* __restrict__ out, int M) {
    const int wave = blockIdx.x * blockDim.y + threadIdx.y;
    const int m0 = wave * 16;
    if (m0 >= M) return;

    const int lane  = threadIdx.x;        // 0..31
    const int l15   = lane & 15;
    const int khalf = (lane >> 4) * 2;    // 0 or 2

    constexpr int NT = NC / 16;
    v8f acc[NT];
#pragma unroll
    for (int t = 0; t < NT; ++t) acc[t] = v8f{0.f, 0.f, 0.f, 0.f, 0.f, 0.f, 0.f, 0.f};

    const float* arow = A + (size_t)(m0 + l15) * GCN_K;

#pragma unroll 4
    for (int k = 0; k < GCN_K; k += 4) {
        const v2f a = *(const v2f*)(arow + k + khalf);  // A[m, k+khalf .. +1]
        const float* w0 = W + (size_t)(k + khalf) * NC + l15;
#pragma unroll
        for (int t = 0; t < NT; ++t) {
            v2f b;
            b.x = w0[t * 16];
            b.y = w0[t * 16 + NC];
            acc[t] = __builtin_amdgcn_wmma_f32_16x16x4_f32(
                false, a, false, b, (short)0, acc[t], false, false);
        }
    }

    const int mbase = m0 + (lane >> 4) * 8;
#pragma unroll
    for (int t = 0; t < NT; ++t) {
        float* orow = out + (size_t)mbase * NC + t * 16 + l15;
#pragma unroll
        for (int r = 0; r < 8; ++r) orow[(size_t)r * NC] = acc[t][r];
    }
}

// ---------------- aggregation ----------------
// B[i,:] = dinv[i]^2 * h[i,:]   (self-loop message, also initializes B)

template <int C>
__global__ void self_init_kernel(const float* __restrict__ h, const float* __restrict__ dinv,
                                 float* __restrict__ out, int n) {
    constexpr int G = C / 4;
    long long tid = (long long)blockIdx.x * blockDim.x + threadIdx.x;
    if (tid >= (long long)n * G) return;
    const int i = (int)(tid / G);
    const int g = (int)(tid % G);
    const float d = dinv[i];
    const float s = d * d;
    const float4 v = *(const float4*)(h + (size_t)i * C + g * 4);
    float4 o = {s * v.x, s * v.y, s * v.z, s * v.w};
    *(float4*)(out + (size_t)i * C + g * 4) = o;
}

// B[col,:] += dinv[row]*dinv[col] * h[row,:]  — C/4 lanes cooperate per edge,
// so row/col/dinv loads are wave-uniform and feature traffic is float4.

template <int C>
__global__ void edge_scatter_kernel(const float* __restrict__ h, const float* __restrict__ dinv,
                                    const int* __restrict__ row, const int* __restrict__ col,
                                    float* __restrict__ out, int e) {
    constexpr int TPE = C / 4;  // 32 for C=128, 16 for C=64
    long long tid = (long long)blockIdx.x * blockDim.x + threadIdx.x;
    const long long ei = tid / TPE;
    const int g = (int)(tid % TPE);
    if (ei >= e) return;
    const int r = row[ei];
    const int c = col[ei];
    const float norm = dinv[r] * dinv[c];
    const float4 v = *(const float4*)(h + (size_t)r * C + g * 4);
    float* dst = out + (size_t)c * C + g * 4;
    atomic_add_f32(dst + 0, norm * v.x);
    atomic_add_f32(dst + 1, norm * v.y);
    atomic_add_f32(dst + 2, norm * v.z);
    atomic_add_f32(dst + 3, norm * v.w);
}

// out = (relu?) (B + bias)

template <int C, bool RELU>
__global__ void bias_act_kernel(const float* __restrict__ in, const float* __restrict__ bias,
                                float* __restrict__ out, int n) {
    constexpr int G = C / 4;
    long long tid = (long long)blockIdx.x * blockDim.x + threadIdx.x;
    if (tid >= (long long)n * G) return;
    const int i = (int)(tid / G);
    const int g = (int)(tid % G);
    const float4 v = *(const float4*)(in + (size_t)i * C + g * 4);
    const float4 b = *(const float4*)(bias + g * 4);
    float4 o = {v.x + b.x, v.y + b.y, v.z + b.z, v.w + b.w};
    if (RELU) {
        o.x = fmaxf(o.x, 0.f); o.y = fmaxf(o.y, 0.f);
        o.z = fmaxf(o.z, 0.f); o.w = fmaxf(o.w, 0.f);
    }
    *(float4*)(out + (size_t)i * C + g * 4) = o;
}

// ---------------- launch ----------------

static inline int cdivll(long long a, int b) { return (int)((a + b - 1) / b); }

extern "C" void kernel_launch(void* const* d_in, const int* in_sizes, int n_in,
                              void* d_out, int out_size, void* d_ws, size_t ws_size,
                              hipStream_t stream) {
    const int N = GCN_N;
    const int E = in_sizes[1] / 2;

    const float* x   = (const float*)d_in[0];
    const int*   ei  = (const int*)d_in[1];
    const int*   row = ei;
    const int*   col = ei + E;
    const float* W1  = (const float*)d_in[2];
    const float* b1  = (const float*)d_in[3];
    const float* W2  = (const float*)d_in[4];
    const float* b2  = (const float*)d_in[5];
    const float* W3  = (const float*)d_in[6];
    const float* b3  = (const float*)d_in[7];

    float* ws   = (float*)d_ws;
    float* dinv = ws;                              // N floats (N % 16 == 0)
    float* A    = dinv + N;                        // N*128: GEMM output (messages)
    float* B    = A + (size_t)N * 128;             // N*128: aggregation target

    float* h2     = (float*)d_out;                 // N*128
    float* logits = h2 + (size_t)N * 128;          // N*64

    const int TB = 256;
    const int mtiles = N / 16;                         // 3125, exact
    const dim3 gblk(32, 4);                            // 4 waves / block
    const int ggrid = (mtiles + 3) / 4;

    // degree -> dinv
    fill_one_kernel<<<cdivll(N, TB), TB, 0, stream>>>(dinv, N);
    degree_scatter_kernel<<<cdivll(E, TB), TB, 0, stream>>>(col, dinv, E);
    rsqrt_kernel<<<cdivll(N, TB), TB, 0, stream>>>(dinv, N);

    // ---- layer 1: h1 = relu(agg(x @ W1) + b1), h1 kept in B ----
    gemm_wmma_f32<128><<<ggrid, gblk, 0, stream>>>(x, W1, A, N);
    self_init_kernel<128><<<cdivll((long long)N * 32, TB), TB, 0, stream>>>(A, dinv, B, N);
    edge_scatter_kernel<128><<<cdivll((long long)E * 32, TB), TB, 0, stream>>>(A, dinv, row, col, B, E);
    bias_act_kernel<128, true><<<cdivll((long long)N * 32, TB), TB, 0, stream>>>(B, b1, B, N);

    // ---- layer 2: h2 = relu(agg(h1 @ W2) + b2) -> d_out[0 : N*128] ----
    gemm_wmma_f32<128><<<ggrid, gblk, 0, stream>>>(B, W2, A, N);
    self_init_kernel<128><<<cdivll((long long)N * 32, TB), TB, 0, stream>>>(A, dinv, B, N);
    edge_scatter_kernel<128><<<cdivll((long long)E * 32, TB), TB, 0, stream>>>(A, dinv, row, col, B, E);
    bias_act_kernel<128, true><<<cdivll((long long)N * 32, TB), TB, 0, stream>>>(B, b2, h2, N);

    // ---- layer 3: out = agg(h2 @ W3) + b3 -> d_out[N*128 : N*192] ----
    gemm_wmma_f32<64><<<ggrid, gblk, 0, stream>>>(h2, W3, A, N);
    self_init_kernel<64><<<cdivll((long long)N * 16, TB), TB, 0, stream>>>(A, dinv, B, N);
    edge_scatter_kernel<64><<<cdivll((long long)E * 16, TB), TB, 0, stream>>>(A, dinv, row, col, B, E);
    bias_act_kernel<64, false><<<cdivll((long long)N * 16, TB), TB, 0, stream>>>(B, b3, logits, N);
}